// Phase2ZModel_27169963114773
// MI455X (gfx1250) — compile-verified
//
#include <hip/hip_runtime.h>
#include <hip/hip_bf16.h>

#define B_ 16
#define T_ 1024
#define H_ 2048
#define KMAX_ 64
#define ZV_ 512
#define NDIG_ 12
#define NCLS_ 10
#define VOCAB_ 32000
#define ANSWER_ID_ 2
#define LATENT_ID_ 3
#define Z_START_ 1000

typedef float v2f __attribute__((ext_vector_type(2)));
typedef float v4f __attribute__((ext_vector_type(4)));
typedef float v8f __attribute__((ext_vector_type(8)));

// ---------------------------------------------------------------------------
// K0: inverse L2 norms of Ez rows (Ez = emb_weight[Z_START : Z_START+ZV])
// ---------------------------------------------------------------------------
__global__ __launch_bounds__(256) void k_ez_norm(const float* __restrict__ emb,
                                                 float* __restrict__ inv_ez) {
    const int z = blockIdx.x;
    const float* row = emb + (size_t)(Z_START_ + z) * H_;
    float ss = 0.f;
    for (int i = threadIdx.x; i < H_; i += 256) {
        float v = row[i];
        ss += v * v;
    }
    __shared__ float red[256];
    red[threadIdx.x] = ss;
    __syncthreads();
    for (int s = 128; s > 0; s >>= 1) {
        if (threadIdx.x < s) red[threadIdx.x] += red[threadIdx.x + s];
        __syncthreads();
    }
    if (threadIdx.x == 0) {
        float n = sqrtf(red[0]);
        inv_ez[z] = 1.0f / fmaxf(n, 1e-12f);
    }
}

// ---------------------------------------------------------------------------
// K1: per-batch scan: build position->k slot map (argsort semantics: latent
// columns ascending, then non-latent ascending, first KMAX, masked by z_mask)
// and the first ANSWER_ID position.
// ---------------------------------------------------------------------------
__global__ void k_scan(const int* __restrict__ ids, const int* __restrict__ zm,
                       int* __restrict__ slot, int* __restrict__ ans_idx) {
    const int b = threadIdx.x;
    if (b >= B_) return;
    const int* r = ids + b * T_;
    int* sl = slot + b * T_;
    for (int t = 0; t < T_; ++t) sl[t] = -1;
    int cnt = 0;
    for (int t = 0; t < T_ && cnt < KMAX_; ++t)
        if (r[t] == LATENT_ID_) {
            if (zm[b * KMAX_ + cnt]) sl[t] = cnt;
            ++cnt;
        }
    for (int t = 0; t < T_ && cnt < KMAX_; ++t)
        if (r[t] != LATENT_ID_) {
            if (zm[b * KMAX_ + cnt]) sl[t] = cnt;
            ++cnt;
        }
    int a = 0;
    for (int t = T_ - 1; t >= 0; --t)
        if (r[t] == ANSWER_ID_) a = t;
    ans_idx[b] = a;
}

// ---------------------------------------------------------------------------
// K2: sim GEMM (1024 x 512 x 2048, f32 WMMA 16x16x4) + row argmax -> z_ids.
// Block = 256 threads (8 waves), one 16-row strip per block; wave w covers
// columns [w*64, w*64+64) as 4 WMMA n-tiles. argmax(sim) == argmin(dist);
// latent-side norm is a per-row positive constant, so only Ez inverse norms
// matter for the argmax.
// ---------------------------------------------------------------------------
__global__ __launch_bounds__(256) void k_sim_argmax(
    const float* __restrict__ ls,      // (B*KMAX, H) latent_states
    const float* __restrict__ emb,     // (VOCAB, H)
    const float* __restrict__ inv_ez,  // (ZV)
    const int* __restrict__ zm,        // (B*KMAX)
    int* __restrict__ z_ids)           // (B*KMAX)
{
    __shared__ float simbuf[16 * ZV_];  // 32 KB
    __shared__ float rv[256];
    __shared__ int ri[256];

    const int m0 = blockIdx.x * 16;
    const int wave = threadIdx.x >> 5;
    const int lane = threadIdx.x & 31;
    const int n0 = wave * 64;
    const int half = lane >> 4;  // 0: lanes 0-15 (K=0,1), 1: lanes 16-31 (K=2,3)
    const int l16 = lane & 15;

    const float* arow = ls + (size_t)(m0 + l16) * H_ + half * 2;
    const float* b0 = emb + (size_t)(Z_START_ + n0 + 0 * 16 + l16) * H_ + half * 2;
    const float* b1 = emb + (size_t)(Z_START_ + n0 + 1 * 16 + l16) * H_ + half * 2;
    const float* b2 = emb + (size_t)(Z_START_ + n0 + 2 * 16 + l16) * H_ + half * 2;
    const float* b3 = emb + (size_t)(Z_START_ + n0 + 3 * 16 + l16) * H_ + half * 2;

    v8f acc0 = {}, acc1 = {}, acc2 = {}, acc3 = {};
    for (int kk = 0; kk < H_; kk += 4) {
        v2f a  = *(const v2f*)(arow + kk);
        v2f v0 = *(const v2f*)(b0 + kk);
        v2f v1 = *(const v2f*)(b1 + kk);
        v2f v2 = *(const v2f*)(b2 + kk);
        v2f v3 = *(const v2f*)(b3 + kk);
        acc0 = __builtin_amdgcn_wmma_f32_16x16x4_f32(false, a, false, v0, (short)0, acc0, false, false);
        acc1 = __builtin_amdgcn_wmma_f32_16x16x4_f32(false, a, false, v1, (short)0, acc1, false, false);
        acc2 = __builtin_amdgcn_wmma_f32_16x16x4_f32(false, a, false, v2, (short)0, acc2, false, false);
        acc3 = __builtin_amdgcn_wmma_f32_16x16x4_f32(false, a, false, v3, (short)0, acc3, false, false);
    }

    // C/D layout: vgpr r, lanes 0-15 -> (M=r, N=lane); lanes 16-31 -> (M=r+8).
    #pragma unroll
    for (int vr = 0; vr < 8; ++vr) {
        const int m = vr + half * 8;
        const int c = n0 + l16;
        simbuf[m * ZV_ + c +  0] = acc0[vr] * inv_ez[c +  0];
        simbuf[m * ZV_ + c + 16] = acc1[vr] * inv_ez[c + 16];
        simbuf[m * ZV_ + c + 32] = acc2[vr] * inv_ez[c + 32];
        simbuf[m * ZV_ + c + 48] = acc3[vr] * inv_ez[c + 48];
    }
    __syncthreads();

    // argmax over z per row (first-max-wins, matching argmin-of-dist)
    const int row = threadIdx.x >> 4;
    const int sub = threadIdx.x & 15;
    float bv = -3.402823e38f;
    int bz = ZV_;
    for (int z = sub; z < ZV_; z += 16) {
        float v = simbuf[row * ZV_ + z];
        if (v > bv || (v == bv && z < bz)) { bv = v; bz = z; }
    }
    rv[threadIdx.x] = bv;
    ri[threadIdx.x] = bz;
    __syncthreads();
    if (sub == 0) {
        for (int s = 1; s < 16; ++s) {
            float v = rv[row * 16 + s];
            int z = ri[row * 16 + s];
            if (v > bv || (v == bv && z < bz)) { bv = v; bz = z; }
        }
        const int r = m0 + row;
        z_ids[r] = zm[r] ? bz : 0;
    }
}

// ---------------------------------------------------------------------------
// K3: big embedding gather with scatter folded in: if position t is a live
// latent slot k, the row index becomes Z_START + z_ids[b,k]. 256 MiB of HBM
// traffic total -> the real cost of this problem (~11 us at 23.3 TB/s).
// Output is written non-temporally (streamed to HBM) so the 192 MB L2 keeps
// the embedding-table rows, which have ~22% duplicate-row reuse.
// ---------------------------------------------------------------------------
__global__ __launch_bounds__(256) void k_gather(
    const int* __restrict__ ids, const float* __restrict__ emb,
    const int* __restrict__ slot, const int* __restrict__ z_ids,
    float* __restrict__ out_embeds) {
    const int bt = blockIdx.x;  // 0 .. B*T-1
    const int b = bt >> 10;
    const int k = __builtin_nontemporal_load(slot + bt);
    const int row = (k >= 0) ? (Z_START_ + z_ids[b * KMAX_ + k]) : ids[bt];
    const v4f* src = (const v4f*)(emb + (size_t)row * H_);
    v4f* dst = (v4f*)(out_embeds + (size_t)bt * H_);
    v4f x0 = src[threadIdx.x];
    v4f x1 = src[threadIdx.x + 256];
    __builtin_nontemporal_store(x0, dst + threadIdx.x);
    __builtin_nontemporal_store(x1, dst + threadIdx.x + 256);
}

// ---------------------------------------------------------------------------
// K4: digit head. One wave per digit d: 16x16x2048 f32 WMMA chain,
// M = batch (16), N = class (10, padded to 16), K = H.
// ---------------------------------------------------------------------------
__global__ __launch_bounds__(32) void k_head(
    const float* __restrict__ embeds, const int* __restrict__ ans_idx,
    const float* __restrict__ head_w, const float* __restrict__ head_b,
    float* __restrict__ logits) {
    const int d = blockIdx.x;
    const int lane = threadIdx.x;
    const int half = lane >> 4;
    const int l16 = lane & 15;

    const float* arow = embeds + ((size_t)l16 * T_ + ans_idx[l16]) * H_ + half * 2;
    const float* wd = head_w + (size_t)d * H_ * NCLS_;
    const int csafe = (l16 < NCLS_) ? l16 : 0;

    v8f acc = {};
    for (int kk = 0; kk < H_; kk += 4) {
        v2f a = *(const v2f*)(arow + kk);
        const int k0 = kk + half * 2;
        float bx = wd[(size_t)k0 * NCLS_ + csafe];
        float by = wd[(size_t)(k0 + 1) * NCLS_ + csafe];
        v2f bb;
        bb.x = (l16 < NCLS_) ? bx : 0.f;
        bb.y = (l16 < NCLS_) ? by : 0.f;
        acc = __builtin_amdgcn_wmma_f32_16x16x4_f32(false, a, false, bb, (short)0, acc, false, false);
    }
    #pragma unroll
    for (int vr = 0; vr < 8; ++vr) {
        const int m = vr + half * 8;  // batch
        if (l16 < NCLS_)
            logits[(m * NDIG_ + d) * NCLS_ + l16] = acc[vr] + head_b[d * NCLS_ + l16];
    }
}

// ---------------------------------------------------------------------------
extern "C" void kernel_launch(void* const* d_in, const int* in_sizes, int n_in,
                              void* d_out, int out_size, void* d_ws, size_t ws_size,
                              hipStream_t stream) {
    const int* ids      = (const int*)d_in[0];
    // d_in[1] = attention_mask (unused by reference math)
    const float* ls     = (const float*)d_in[2];
    const int* zm       = (const int*)d_in[3];
    const float* emb    = (const float*)d_in[4];
    const float* head_w = (const float*)d_in[5];
    const float* head_b = (const float*)d_in[6];

    float* logits     = (float*)d_out;                        // (B, NDIG, NCLS)
    float* out_embeds = (float*)d_out + B_ * NDIG_ * NCLS_;   // (B, T, H)

    // workspace layout
    float* inv_ez = (float*)d_ws;                 // 512 f32
    int* z_ids    = (int*)d_ws + 512;             // 1024 i32
    int* slot     = (int*)d_ws + 1536;            // B*T i32
    int* ans      = (int*)d_ws + 1536 + B_ * T_;  // 16 i32

    k_ez_norm<<<ZV_, 256, 0, stream>>>(emb, inv_ez);
    k_scan<<<1, 32, 0, stream>>>(ids, zm, slot, ans);
    k_sim_argmax<<<(B_ * KMAX_) / 16, 256, 0, stream>>>(ls, emb, inv_ez, zm, z_ids);
    k_gather<<<B_ * T_, 256, 0, stream>>>(ids, emb, slot, z_ids, out_embeds);
    k_head<<<NDIG_, 32, 0, stream>>>(out_embeds, ans, head_w, head_b, logits);
}